// WeightedCrossEntropyLoss_69483980915154
// MI455X (gfx1250) — compile-verified
//
#include <hip/hip_runtime.h>
#include <hip/hip_bf16.h>

typedef __attribute__((ext_vector_type(2))) float v2f;
typedef __attribute__((ext_vector_type(8))) float v8f;

#define BB   16
#define CC   8
#define HW   (512 * 512)          // 2^18
#define NPIX (BB * HW)            // 4,194,304
#define GRID_BLOCKS 2048          // 2048*256 threads -> exactly 8 pixels/thread
                                  // (max per-thread class count 8 << 255, so
                                  //  per-class counts pack into one byte each)

// ---------------------------------------------------------------------------
// Zero the 16 global accumulators (8 class loss sums + 8 class counts).
// ---------------------------------------------------------------------------
__global__ void wce_init(float* __restrict__ accum) {
    if (threadIdx.x < 16) accum[threadIdx.x] = 0.0f;
}

// ---------------------------------------------------------------------------
// Main pass: streaming log-softmax CE per pixel, per-lane class bins,
// block reduction via V_WMMA_F32_16X16X4_F32 (ones x data = column sums),
// global_atomic_add_f32 into accum[0..15] = {sums[8], counts[8]}.
// ---------------------------------------------------------------------------
__global__ __launch_bounds__(256) void wce_main(const float* __restrict__ y_pred,
                                                const int*   __restrict__ y_true,
                                                float*       __restrict__ accum) {
    __shared__ float lds[256 * 16];   // 16 KB: row = thread, cols 0-7 sums, 8-15 counts

    float s[8];
#pragma unroll
    for (int c = 0; c < 8; ++c) s[c] = 0.0f;
    unsigned long long cnt64 = 0ull;  // 8 packed byte counters (<=8 pixels/thread)

    const int stride = gridDim.x * blockDim.x;
    for (int p = blockIdx.x * blockDim.x + threadIdx.x; p < NPIX; p += stride) {
        const int b    = p >> 18;                 // HW = 2^18
        const int base = p + b * (7 * HW);        // = b*C*HW + hw  (fits in int)

        // Keep the HBM stream ahead of the VALU work (global_prefetch_b8).
        const int pn = p + stride;
        if (pn < NPIX) {
            const int bn = pn >> 18;
            __builtin_prefetch(&y_pred[pn + bn * (7 * HW)], 0, 1);
            __builtin_prefetch(&y_true[pn], 0, 1);
        }

        float l[8];
#pragma unroll
        for (int c = 0; c < 8; ++c) l[c] = y_pred[base + c * HW];   // coalesced per class plane

        float m = l[0];
#pragma unroll
        for (int c = 1; c < 8; ++c) m = fmaxf(m, l[c]);
        float sum = 0.0f;
#pragma unroll
        for (int c = 0; c < 8; ++c) sum += __expf(l[c] - m);

        const int lab = y_true[p] & 7;
        const float lse = __logf(sum) + m;        // per-pixel loss = lse - l[lab]

        // Static-index binning: no dynamic l[lab] extraction chain.
#pragma unroll
        for (int c = 0; c < 8; ++c)
            s[c] += (c == lab) ? (lse - l[c]) : 0.0f;

        cnt64 += 1ull << (lab << 3);              // packed per-class count
    }

    const int tid = threadIdx.x;
#pragma unroll
    for (int c = 0; c < 8; ++c) {
        lds[tid * 16 + c]     = s[c];
        lds[tid * 16 + 8 + c] = (float)((cnt64 >> (c << 3)) & 0xffull);
    }
    __syncthreads();

    // Each wave column-sums its own 32x16 slice of LDS with 8 chained f32 WMMAs.
    // D = sum_k ones(16x4) x B(4x16); with A==1 the row->K mapping inside each
    // 4-row chunk is irrelevant, so only the documented lane&15 -> N column
    // mapping matters. All 32 lanes of every wave are active (EXEC all ones).
    const int wave = tid >> 5;
    const int lane = tid & 31;
    const int col  = lane & 15;
    const int rsel = (lane >> 4) << 1;            // half-wave row select: 0 or 2
    const int rowbase = wave * 32;

    v8f acc = {};
    v2f a;  a.x = 1.0f;  a.y = 1.0f;              // all-ones A matrix (layout invariant)
#pragma unroll
    for (int k = 0; k < 8; ++k) {
        const int r0 = rowbase + 4 * k + rsel;
        v2f bm;
        bm.x = lds[r0 * 16 + col];
        bm.y = lds[(r0 + 1) * 16 + col];
        acc = __builtin_amdgcn_wmma_f32_16x16x4_f32(
                  /*neg_a=*/false, a, /*neg_b=*/false, bm,
                  /*c_mod=*/(short)0, acc, /*reuse_a=*/false, /*reuse_b=*/false);
    }

    // D row M=0 lives in acc[0] of lanes 0..15 (N = lane): the 16 column totals.
    if (lane < 16) atomicAdd(&accum[lane], acc[0]);
}

// ---------------------------------------------------------------------------
// Finalize: normalize weights, guarded per-class mean, weighted sum -> scalar.
// ---------------------------------------------------------------------------
__global__ void wce_final(const float* __restrict__ wts,
                          const float* __restrict__ accum,
                          float*       __restrict__ out) {
    if (threadIdx.x == 0) {
        float wsum = 0.0f;
        for (int c = 0; c < 8; ++c) wsum += wts[c];
        float res = 0.0f;
        for (int c = 0; c < 8; ++c) {
            const float sc = accum[c];
            const float cc = accum[8 + c];
            const float mean = (cc > 0.0f) ? sc / fmaxf(cc, 1.0f) : 0.0f;
            res += (wts[c] / wsum) * mean;
        }
        out[0] = res;
    }
}

extern "C" void kernel_launch(void* const* d_in, const int* in_sizes, int n_in,
                              void* d_out, int out_size, void* d_ws, size_t ws_size,
                              hipStream_t stream) {
    const float* y_pred = (const float*)d_in[0];   // [16,8,512,512] f32
    const float* wts    = (const float*)d_in[1];   // [8] f32 (un-normalized)
    const int*   y_true = (const int*)d_in[2];     // [16,512,512] labels
    float* accum = (float*)d_ws;                   // 16 floats: sums[8] || counts[8]
    float* out   = (float*)d_out;                  // scalar

    wce_init<<<1, 32, 0, stream>>>(accum);
    wce_main<<<GRID_BLOCKS, 256, 0, stream>>>(y_pred, y_true, accum);
    wce_final<<<1, 32, 0, stream>>>(wts, accum, out);
}